// Embedding_44555990729103
// MI455X (gfx1250) — compile-verified
//
#include <hip/hip_runtime.h>
#include <hip/hip_bf16.h>

#define THREADS 256
#define ROWS_PER_BLOCK 64   // 64 rows * 64 cols * 4B = 16 KB contiguous input slab
#define XCOLS 64
#define OCOLS 80

// Packed per-output-column map (slow cols 0..27): kind(2b) | xcol<<2 | woff<<8
//   kind 0: out = x[row, xcol]
//   kind 1: out = W2[(int)x[row,xcol], woff]   (W2 flat [2,18])
//   kind 2: out = W3[(int)x[row,6],    woff]   (W3 flat [3,5])
__constant__ int g_colmap[OCOLS] = {
       1,  257,  513,    4,  777, 1033, 1289,   12, 1553, 1809,
    2065,   20,   26,  282,  538,  794, 1050, 2333, 2589, 2845,
      32, 3109, 3365, 3621,   40, 3885, 4141, 4397,   48,   52,
      56,   60,   64,   68,   72,   76,   80,   84,   88,   92,
      96,  100,  104,  108,  112,  116,  120,  124,  128,  132,
     136,  140,  144,  148,  152,  156,  160,  164,  168,  172,
     176,  180,  184,  188,  192,  196,  200,  204,  208,  212,
     216,  220,  224,  228,  232,  236,  240,  244,  248,  252
};

__device__ __forceinline__ unsigned lds_off32(const void* p) {
    // For the LDS aperture, flat_addr[31:0] is the LDS byte offset.
    return (unsigned)(unsigned long long)p;
}

typedef unsigned int u32x4 __attribute__((ext_vector_type(4)));
typedef int          i32x4 __attribute__((ext_vector_type(4)));
typedef int          i32x8 __attribute__((ext_vector_type(8)));

__global__ __launch_bounds__(THREADS)
void emb_tile_kernel(const float* __restrict__ x,
                     const float* __restrict__ W2,
                     const float* __restrict__ W3,
                     float* __restrict__ out)
{
    __shared__ alignas(16) float tile[ROWS_PER_BLOCK * XCOLS]; // 16 KB
    __shared__ float ws[64];     // [0..35]=W2 flat, [36..50]=W3 flat, rest pad
    __shared__ int   cmap[28];   // slow-path column map

    const int tid = threadIdx.x;
    const long long row0 = (long long)blockIdx.x * ROWS_PER_BLOCK;

    // ---- stage tiny tables
    if (tid < 36)       ws[tid] = W2[tid];
    else if (tid < 51)  ws[tid] = W3[tid - 36];
    else if (tid < 64)  ws[tid] = 0.0f;
    if (tid < 28)       cmap[tid] = g_colmap[tid];

    // ---- stage the 64x64xfp32 slab of x into LDS
    const float* src = x + row0 * XCOLS;
#if defined(__HIP_DEVICE_COMPILE__) && __has_builtin(__builtin_amdgcn_tensor_load_to_lds)
    // CDNA5 Tensor Data Mover: one DMA for the whole 16 KB tile (wave 0 issues).
    if (tid < 32) {
        unsigned long long ga = (unsigned long long)src;
        u32x4 g0;
        g0.x = 1u;                                   // count=1, user mode, no gather
        g0.y = lds_off32(&tile[0]);                  // lds_addr
        g0.z = (unsigned)(ga & 0xffffffffu);         // global_addr[31:0]
        g0.w = (unsigned)((ga >> 32) & 0x01ffffffu)  // global_addr[56:32]
             | 0x80000000u;                          // type=2 ("image")
        i32x8 g1;
        g1[0] = 0x00020000;     // data_size=2 (4B), mask=0, no iterate/pad/atomic
        g1[1] = (int)(64u << 16);   // tensor_dim0[15:0]=64
        g1[2] = (int)(64u << 16);   // tensor_dim1[15:0]=64 (dim0 hi=0)
        g1[3] = (int)(64u << 16);   // tile_dim0=64       (dim1 hi=0)
        g1[4] = 64;                 // tile_dim1=64, tile_dim2=0
        g1[5] = 64;                 // tensor_dim0_stride = 64 elements
        g1[6] = 0;                  // stride0 hi, stride1 lo (unused, 2D)
        g1[7] = 0;
        i32x4 z4 = {0, 0, 0, 0};
#if __has_include(<hip/amd_detail/amd_gfx1250_TDM.h>)
        i32x8 z8 = {0, 0, 0, 0, 0, 0, 0, 0};
        __builtin_amdgcn_tensor_load_to_lds(g0, g1, z4, z4, z8, 0);
#else
        __builtin_amdgcn_tensor_load_to_lds(g0, g1, z4, z4, 0);
#endif
        __builtin_amdgcn_s_wait_tensorcnt(0);
    }
#else
    // Fallback: per-lane async global->LDS b128 copies (ASYNCcnt path).
    {
        const unsigned lds_base = lds_off32(&tile[0]);
        #pragma unroll
        for (int i = 0; i < (ROWS_PER_BLOCK * XCOLS / 4) / THREADS; ++i) { // 4
            unsigned f4   = (unsigned)(i * THREADS + tid);
            unsigned goff = f4 * 16u;
            unsigned loff = lds_base + goff;
            asm volatile("global_load_async_to_lds_b128 %0, %1, %2"
                         :: "v"(loff), "v"(goff), "s"(src)
                         : "memory");
        }
        asm volatile("s_wait_asynccnt 0" ::: "memory");
    }
#endif
    __syncthreads();

    float* dst = out + row0 * OCOLS;

    // ---- Loop A: pass-through chunks (out cols 28..79 <- x cols 12..63)
    // 64 rows * 13 chunks = 832 float4 chunks; pure b128 copy, no decode.
    #pragma unroll
    for (int i = 0; i < 4; ++i) {
        const int f = i * THREADS + tid;
        if (f < ROWS_PER_BLOCK * 13) {
            const int r  = f / 13;
            const int cj = f - r * 13;
            const float4 v = *(const float4*)(tile + r * XCOLS + 12 + 4 * cj);
            *(float4*)(dst + r * OCOLS + 28 + 4 * cj) = v;
        }
    }

    // ---- Loop B: gather chunks (out cols 0..27), branchless select
    // 64 rows * 7 chunks = 448 float4 chunks.
    #pragma unroll
    for (int i = 0; i < 2; ++i) {
        const int f = i * THREADS + tid;
        if (f < ROWS_PER_BLOCK * 7) {
            const int r  = f / 7;
            const int c0 = 4 * (f - r * 7);
            const float* xr = &tile[r * XCOLS];
            float t[4];
            #pragma unroll
            for (int k = 0; k < 4; ++k) {
                const int e    = cmap[c0 + k];
                const int kind = e & 3;
                const int xcol = (e >> 2) & 63;
                const int woff = e >> 8;
                const float xv = xr[xcol];
                int idx = __float2int_rz(xv);
                idx = idx < 0 ? 0 : idx;
                const int i2 = idx > 1 ? 1 : idx;       // clamp for W2 (2 rows)
                const int i3 = idx > 2 ? 2 : idx;       // clamp for W3 (3 rows)
                const float w2v = ws[i2 * 18 + woff];        // <= 53 < 64, in-bounds
                const float w3v = ws[36 + i3 * 5 + woff];    // <= 63, in-bounds
                t[k] = (kind == 0) ? xv : ((kind == 1) ? w2v : w3v);
            }
            *(float4*)(dst + r * OCOLS + c0) = make_float4(t[0], t[1], t[2], t[3]);
        }
    }
}

// Tail: one thread per output element for the last (B % 64) rows.
__global__ __launch_bounds__(THREADS)
void emb_tail_kernel(const float* __restrict__ x,
                     const float* __restrict__ W2,
                     const float* __restrict__ W3,
                     float* __restrict__ out,
                     long long row_start, long long nrows)
{
    long long g = (long long)blockIdx.x * blockDim.x + threadIdx.x;
    long long total = nrows * OCOLS;
    if (g >= total) return;
    long long r = row_start + g / OCOLS;
    int c = (int)(g % OCOLS);
    const int e    = g_colmap[c];
    const int kind = e & 3;
    const int xcol = (e >> 2) & 63;
    const int woff = e >> 8;
    const float xv = x[r * XCOLS + xcol];
    float val;
    if (kind == 0)      val = xv;
    else if (kind == 1) val = W2[__float2int_rz(xv) * 18 + woff];
    else                val = W3[__float2int_rz(xv) * 5  + woff];
    out[r * OCOLS + c] = val;
}

extern "C" void kernel_launch(void* const* d_in, const int* in_sizes, int n_in,
                              void* d_out, int out_size, void* d_ws, size_t ws_size,
                              hipStream_t stream) {
    const float* x  = (const float*)d_in[0];
    const float* W2 = (const float*)d_in[1];
    const float* W3 = (const float*)d_in[2];
    float* out = (float*)d_out;

    const long long B     = (long long)in_sizes[0] / XCOLS;   // 1,000,000
    const long long fullB = B / ROWS_PER_BLOCK;               // full 64-row tiles
    const long long rem   = B - fullB * ROWS_PER_BLOCK;

    if (fullB > 0) {
        emb_tile_kernel<<<(unsigned)fullB, THREADS, 0, stream>>>(x, W2, W3, out);
    }
    if (rem > 0) {
        long long elems = rem * OCOLS;
        unsigned blocks = (unsigned)((elems + THREADS - 1) / THREADS);
        emb_tail_kernel<<<blocks, THREADS, 0, stream>>>(x, W2, W3, out,
                                                        fullB * ROWS_PER_BLOCK, rem);
    }
}